// DistanceMatrixRaggedModel_88622355186251
// MI455X (gfx1250) — compile-verified
//
#include <hip/hip_runtime.h>
#include <hip/hip_bf16.h>

// Problem constants (from reference)
#define MAXN   512
#define DFEAT  (MAXN * MAXN)   // 262144
#define BATCH  256
#define HIDDEN 128
#define PHI    128
#define RHO1   256
#define RHO2   128
#define OUTF   64

#define NTW    4               // N-tiles (of 16) per wave -> 16x64 output/wave
#define NGROUP (HIDDEN / (16 * NTW))   // = 2 N-groups

typedef __attribute__((ext_vector_type(2))) float v2f;
typedef __attribute__((ext_vector_type(8))) float v8f;

// ---------------------------------------------------------------------------
// Phase 1: partial[ks][b][n] = sum_{k in chunk} mask(x)[b,k] * W1[k,n].
// One wave per (ks, mtile, ngroup); wave computes a 16x64 fp32 tile with
// 4 accumulators, sharing one masked A fragment across 4 V_WMMA_F32_16X16X4.
// ---------------------------------------------------------------------------
__global__ __launch_bounds__(32) void phase1_gemm_wmma(
    const float* __restrict__ x,      // [BATCH, MAXN, MAXN]
    const float* __restrict__ W1,     // [DFEAT, HIDDEN]
    const int*   __restrict__ sizes,  // [BATCH]
    float*       __restrict__ partial,
    int rowsPerChunk)                 // = MAXN / KSPLIT
{
    const int lane   = threadIdx.x;    // 0..31 (wave32)
    const int half   = lane >> 4;      // 0 or 1
    const int lanelo = lane & 15;

    const int ks  = blockIdx.x >> 5;        // K-chunk id
    const int rem = blockIdx.x & 31;
    const int mt  = rem >> 1;               // 16 M tiles
    const int ng  = rem & (NGROUP - 1);     // 2 N groups of 64 cols
    const int m0  = mt * 16;
    const int n0  = ng * (16 * NTW);

    const int mrow = m0 + lanelo;      // batch row this lane feeds into A
    const int s    = sizes[mrow];      // per-lane ragged size

    // wave-uniform max size over this M tile (enables uniform skipping)
    int smax = 0;
#pragma unroll
    for (int t = 0; t < 16; ++t) smax = max(smax, sizes[m0 + t]);

    const float* xrow  = x  + (size_t)mrow * DFEAT;
    const float* wbase = W1 + n0 + lanelo;      // this lane's B column base

    v8f acc0 = {0.f,0.f,0.f,0.f,0.f,0.f,0.f,0.f};
    v8f acc1 = acc0, acc2 = acc0, acc3 = acc0;

    const int i0   = ks * rowsPerChunk;
    const int ilim = min(i0 + rowsPerChunk, smax);   // rows >= smax contribute 0
    const int jmax = min(MAXN, (smax + 3) & ~3);     // cols >= smax contribute 0

    const int sbase = s - 2 * half;    // per-lane col limit base for this half

    for (int i = i0; i < ilim; ++i) {
        const bool rowok = (i < s);
        const int  lim0  = rowok ? sbase     : 0;    // a.x valid iff j < lim0
        const int  lim1  = rowok ? sbase - 1 : 0;    // a.y valid iff j < lim1

        const float* xp = xrow + (size_t)i * MAXN + 2 * half;
        const float* wp = wbase + (size_t)i * MAXN * HIDDEN
                                + (size_t)(2 * half) * HIDDEN;

#pragma unroll 2
        for (int j = 0; j < jmax; j += 4) {
            // A fragment (shared by all 4 N-tiles): lane holds K = j+2h, j+2h+1
            const float2 av = *reinterpret_cast<const float2*>(xp);
            v2f a;
            a.x = (j < lim0) ? av.x : 0.0f;
            a.y = (j < lim1) ? av.y : 0.0f;

            // B fragments: rows K = j+2h, j+2h+1; cols n0 + 16*tn + lanelo
            v2f b0, b1, b2, b3;
            b0.x = wp[ 0];  b0.y = wp[HIDDEN +  0];
            b1.x = wp[16];  b1.y = wp[HIDDEN + 16];
            b2.x = wp[32];  b2.y = wp[HIDDEN + 32];
            b3.x = wp[48];  b3.y = wp[HIDDEN + 48];

            acc0 = __builtin_amdgcn_wmma_f32_16x16x4_f32(false, a, false, b0, (short)0, acc0, false, false);
            acc1 = __builtin_amdgcn_wmma_f32_16x16x4_f32(false, a, false, b1, (short)0, acc1, false, false);
            acc2 = __builtin_amdgcn_wmma_f32_16x16x4_f32(false, a, false, b2, (short)0, acc2, false, false);
            acc3 = __builtin_amdgcn_wmma_f32_16x16x4_f32(false, a, false, b3, (short)0, acc3, false, false);

            xp += 4;
            wp += 4 * HIDDEN;
        }
    }

    // C layout: VGPR v, lane half h -> row m = v + 8*h, col n = lanelo
    float* pout = partial + ((size_t)ks * BATCH + m0) * HIDDEN + n0 + lanelo;
#pragma unroll
    for (int v = 0; v < 8; ++v) {
        const size_t r = (size_t)(v + 8 * half) * HIDDEN;
        pout[r +  0] = acc0[v];
        pout[r + 16] = acc1[v];
        pout[r + 32] = acc2[v];
        pout[r + 48] = acc3[v];
    }
}

// ---------------------------------------------------------------------------
// Phase 2: fixed-order K-split reduction + bias + ReLU, then the tiny tail MLP
// (128->128->256->128->64) per batch row, staged in LDS. Deterministic.
// ---------------------------------------------------------------------------
__global__ __launch_bounds__(256) void phase2_mlp(
    const float* __restrict__ partial,
    const float* __restrict__ b1,
    const float* __restrict__ W2,  const float* __restrict__ b2,
    const float* __restrict__ Wr1, const float* __restrict__ br1,
    const float* __restrict__ Wr2, const float* __restrict__ br2,
    const float* __restrict__ Wr3, const float* __restrict__ br3,
    float*       __restrict__ out,
    int ksplit)
{
    __shared__ float h1[HIDDEN];
    __shared__ float h2[PHI];
    __shared__ float h3[RHO1];
    __shared__ float h4[RHO2];

    const int b = blockIdx.x;
    const int t = threadIdx.x;

    if (t < HIDDEN) {
        float s = b1[t];
        const float* p = partial + (size_t)b * HIDDEN + t;
        for (int k = 0; k < ksplit; ++k)
            s += p[(size_t)k * BATCH * HIDDEN];
        h1[t] = fmaxf(s, 0.0f);
    }
    __syncthreads();

    if (t < PHI) {
        float s = b2[t];
        for (int k = 0; k < HIDDEN; ++k)
            s = fmaf(h1[k], W2[k * PHI + t], s);
        h2[t] = fmaxf(s, 0.0f);
    }
    __syncthreads();

    {   // RHO1 = 256 outputs, one per thread
        float s = br1[t];
        for (int k = 0; k < PHI; ++k)
            s = fmaf(h2[k], Wr1[k * RHO1 + t], s);
        h3[t] = fmaxf(s, 0.0f);
    }
    __syncthreads();

    if (t < RHO2) {
        float s = br2[t];
        for (int k = 0; k < RHO1; ++k)
            s = fmaf(h3[k], Wr2[k * RHO2 + t], s);
        h4[t] = fmaxf(s, 0.0f);
    }
    __syncthreads();

    if (t < OUTF) {
        float s = br3[t];
        for (int k = 0; k < RHO2; ++k)
            s = fmaf(h4[k], Wr3[k * OUTF + t], s);
        out[(size_t)b * OUTF + t] = s;
    }
}

// ---------------------------------------------------------------------------
extern "C" void kernel_launch(void* const* d_in, const int* in_sizes, int n_in,
                              void* d_out, int out_size, void* d_ws, size_t ws_size,
                              hipStream_t stream) {
    const float* x    = (const float*)d_in[0];
    const float* W1   = (const float*)d_in[1];
    const float* b1   = (const float*)d_in[2];
    const float* W2   = (const float*)d_in[3];
    const float* b2   = (const float*)d_in[4];
    const float* Wr1  = (const float*)d_in[5];
    const float* br1  = (const float*)d_in[6];
    const float* Wr2  = (const float*)d_in[7];
    const float* br2  = (const float*)d_in[8];
    const float* Wr3  = (const float*)d_in[9];
    const float* br3  = (const float*)d_in[10];
    const int*   sz   = (const int*)d_in[11];
    float*       out  = (float*)d_out;
    float*       part = (float*)d_ws;

    // K-split for parallelism: partials need ksplit * 256 * 128 * 4 bytes.
    int ksplit = 64;
    while (ksplit > 1 &&
           (size_t)ksplit * BATCH * HIDDEN * sizeof(float) > ws_size)
        ksplit >>= 1;
    const int rowsPerChunk = MAXN / ksplit;

    dim3 grid1(16 * NGROUP * ksplit);   // 16 M-tiles * 2 N-groups * ksplit
    phase1_gemm_wmma<<<grid1, 32, 0, stream>>>(x, W1, sz, part, rowsPerChunk);

    phase2_mlp<<<BATCH, 256, 0, stream>>>(part, b1, W2, b2, Wr1, br1,
                                          Wr2, br2, Wr3, br3, out, ksplit);
}